// Clustering_42348377539001
// MI455X (gfx1250) — compile-verified
//
#include <hip/hip_runtime.h>
#include <math.h>

// ---------------------------------------------------------------------------
// MI455X plan: problem is tiny (≈1 GFLOP GEMM, ≤4MB per tensor, all L2-
// resident at 192MB). Keep f32 end-to-end (reference is f32; softmax/argmax
// are tie-sensitive). Matrix path: fused X@[W_k|W_q|0] as a 512x16 GEMM using
// V_WMMA_F32_16X16X4_F32, one wave32 per 16-row tile, B staged in LDS.
// Inner loop: incremental address gen (no divides), branchless clamped
// gathers (+cndmask zero) with scalar-base + u32 offset (saddr form),
// software-pipelined loads ahead of each WMMA.
// ---------------------------------------------------------------------------

typedef float v2f __attribute__((ext_vector_type(2)));
typedef float v8f __attribute__((ext_vector_type(8)));

#define BB 8
#define HH 8
#define LL 256
#define DKK 64
#define CCL 5
#define DM 512
#define UU 50
#define ROWS_PER_B (LL * UU)       // 12800
#define NROWS (BB * ROWS_PER_B)    // 102400
#define OUT_ELTS (BB * HH * LL * DKK) // 1048576

// workspace layout (float offsets)
#define QOFF   0
#define MUOFF  (NROWS * CCL)            // 512000
#define LPOFF  (MUOFF + NROWS)          // 614400
#define INDOFF (LPOFF + NROWS)          // 716800
#define CTOFF  (INDOFF + NROWS)         // 819200
#define ACCOFF (CTOFF + CCL * BB * LL * CCL) // 870400

__device__ __forceinline__ float gelu_exact(float x) {
    return 0.5f * x * (1.0f + erff(x * 0.70710678118654752440f));
}

// Branchless X gather: element is K[(b+u-49), h, l, d] when b+u>=50 else 0.
// K stays the uniform (scalar) base so the load uses the saddr+voffset form;
// hbase = h*16384, off = l*64 + d (contiguous across the d->l carry).
__device__ __forceinline__ float loadXe(const float* __restrict__ K,
                                        unsigned hbase, int b, int u, int off) {
    int t = b + u;                 // [0,56]
    int ti = t - 49;               // [-49,7]
    int tc = ti > 0 ? ti : 1;      // clamped -> always a valid K row
    unsigned idx = (unsigned)tc * 131072u + hbase + (unsigned)off;
    float v = K[idx];
    return t >= 50 ? v : 0.0f;
}

// ---------------------------------------------------------------------------
// Kernel 1: WMMA GEMM X[102400,512] @ Wcat[512,16] (cols 0-4=W_k, 5-9=W_q),
// then per-row gelu+softmax, stats, argmax. 4 waves/block, 1 tile/wave.
// ---------------------------------------------------------------------------
__global__ __launch_bounds__(128) void cluster_kernel(
    const float* __restrict__ K,
    const float* __restrict__ Wk, const float* __restrict__ bk,
    const float* __restrict__ Wq, const float* __restrict__ bq,
    float* __restrict__ ws)
{
    // Wcat pair-interleaved: addr(k,n) = (k>>1)*32 + n*2 + (k&1)
    // -> each lane's two B values are one ds_load_b64; 64 words/wave = 64 banks.
    __shared__ float lds_w[DM * 16];   // 32KB
    __shared__ float lds_d[4][256];    // per-wave 16x16 D tile

    int tid = threadIdx.x;
    for (int i = tid; i < DM * 16; i += 128) {
        int k = i >> 4, n = i & 15;
        float v = 0.0f;
        if (n < 5)       v = Wk[k * 5 + n];
        else if (n < 10) v = Wq[k * 5 + (n - 5)];
        lds_w[(k >> 1) * 32 + n * 2 + (k & 1)] = v;
    }
    __syncthreads();

    int wv = tid >> 5;
    int lane = tid & 31;
    int n16 = lane & 15;
    int khalf = lane >> 4;
    int tile = blockIdx.x * 4 + wv;    // 6400 tiles total

    // A-operand row for this lane: global row R, decode (b, h, r2)
    int R = tile * 16 + n16;
    int b = R / ROWS_PER_B;            // uniform across the wave's tile
    int rr = R - b * ROWS_PER_B;
    int h = rr / 1600;
    int r2 = rr - h * 1600;
    unsigned hbase = (unsigned)h * 16384u;

    v8f acc = {0.f, 0.f, 0.f, 0.f, 0.f, 0.f, 0.f, 0.f};

    if (b != 0) {   // batch 0: X is identically zero -> acc stays 0
        // prologue: decode once, then advance incrementally (u wrap => off++)
        int cini = 2 * khalf;
        int rem1 = r2 * 512 + cini;
        int l0   = rem1 / 3200;
        int rem2 = rem1 - l0 * 3200;
        int dd   = rem2 / 50;
        int u0   = rem2 - dd * 50;
        int off0 = l0 * 64 + dd;

        float a0 = loadXe(K, hbase, b, u0, off0);
        int uu = u0 + 1, oo = off0;
        if (uu == 50) { uu = 0; ++oo; }
        float a1 = loadXe(K, hbase, b, uu, oo);
        v2f Bv = *(const v2f*)&lds_w[khalf * 32 + n16 * 2];

        #pragma unroll 4
        for (int k0 = 4; k0 < DM; k0 += 4) {
            v2f Acur; Acur.x = a0; Acur.y = a1;
            v2f Bcur = Bv;
            // advance lane's c by 4 and prefetch next step
            u0 += 4;
            int wrap = (u0 >= 50);
            u0 = wrap ? u0 - 50 : u0;
            off0 += wrap;
            a0 = loadXe(K, hbase, b, u0, off0);
            int u1 = u0 + 1, o1 = off0;
            if (u1 == 50) { u1 = 0; ++o1; }
            a1 = loadXe(K, hbase, b, u1, o1);
            Bv = *(const v2f*)&lds_w[((k0 >> 1) + khalf) * 32 + n16 * 2];

            acc = __builtin_amdgcn_wmma_f32_16x16x4_f32(
                false, Acur, false, Bcur, (short)0, acc, false, false);
        }
        v2f Alast; Alast.x = a0; Alast.y = a1;
        acc = __builtin_amdgcn_wmma_f32_16x16x4_f32(
            false, Alast, false, Bv, (short)0, acc, false, false);
    }

    // D layout: lane holds col N=n16, rows M=8*khalf+j
    #pragma unroll
    for (int j = 0; j < 8; ++j)
        lds_d[wv][(8 * khalf + j) * 16 + n16] = acc[j];
    __syncthreads();

    if (lane < 16) {
        int Rr = tile * 16 + lane;
        float gk[5], gq[5];
        #pragma unroll
        for (int j = 0; j < 5; ++j) {
            gk[j] = gelu_exact(lds_d[wv][lane * 16 + j]     + bk[j]);
            gq[j] = gelu_exact(lds_d[wv][lane * 16 + 5 + j] + bq[j]);
        }
        float mk = gk[0], mq = gq[0];
        #pragma unroll
        for (int j = 1; j < 5; ++j) { mk = fmaxf(mk, gk[j]); mq = fmaxf(mq, gq[j]); }
        float sk = 0.f, sq = 0.f;
        #pragma unroll
        for (int j = 0; j < 5; ++j) {
            gk[j] = expf(gk[j] - mk); sk += gk[j];
            gq[j] = expf(gq[j] - mq); sq += gq[j];
        }
        float qk[5], qq[5];
        #pragma unroll
        for (int j = 0; j < 5; ++j) { qk[j] = gk[j] / sk; qq[j] = gq[j] / sq; }

        float sumq = 0.f, sumk = 0.f;
        #pragma unroll
        for (int j = 0; j < 5; ++j) { sumq += qq[j]; sumk += qk[j]; }
        float mu = sumq * 0.2f;
        float xk = sumk * 0.2f;
        float var = 0.f;
        #pragma unroll
        for (int j = 0; j < 5; ++j) { float d0 = qq[j] - mu; var += d0 * d0; }
        var *= 0.25f;                               // ddof=1, n=5
        float stdq = sqrtf(var);
        float sigma = log1pf(expf(stdq));           // softplus
        float z = (xk - mu) / sigma;
        float logp = -0.5f * z * z - logf(sigma) - 0.91893853320467274178f;

        int amax = 0; float bq0 = qq[0];
        #pragma unroll
        for (int j = 1; j < 5; ++j) if (qq[j] > bq0) { bq0 = qq[j]; amax = j; }

        float* qbuf = ws + QOFF;
        #pragma unroll
        for (int j = 0; j < 5; ++j) qbuf[Rr * 5 + j] = qq[j];
        ws[MUOFF + Rr] = mu;
        ws[LPOFF + Rr] = logp;
        ((int*)(ws + INDOFF))[Rr] = amax;
    }
}

// ---------------------------------------------------------------------------
// Kernel 2: cluster centers per (b,l): centers[c][j] = (S[j]-T[c][j])/50
// ---------------------------------------------------------------------------
__global__ __launch_bounds__(256) void centers_kernel(float* __restrict__ ws)
{
    int gid = blockIdx.x * 256 + threadIdx.x;
    if (gid >= BB * LL) return;
    int bb = gid >> 8;
    int l  = gid & 255;
    const float* qbuf = ws + QOFF;
    const int* indp = (const int*)(ws + INDOFF);
    float S[5] = {0, 0, 0, 0, 0};
    float T[25] = {0};
    int base = bb * ROWS_PER_B + l * UU;
    for (int u = 0; u < UU; ++u) {
        int Rr = base + u;
        int id = indp[Rr];
        #pragma unroll
        for (int j = 0; j < 5; ++j) {
            float v = qbuf[Rr * 5 + j];
            S[j] += v;
            T[id * 5 + j] += v;
        }
    }
    float* ct = ws + CTOFF;
    #pragma unroll
    for (int c = 0; c < 5; ++c)
        #pragma unroll
        for (int j = 0; j < 5; ++j)
            ct[((c * BB + bb) * LL + l) * 5 + j] = (S[j] - T[c * 5 + j]) * 0.02f;
}

// ---------------------------------------------------------------------------
// Kernel 3: fused back-projection + gelu + faithful-reshape decode + max_c
// out[b',h,l',d] = max_{c'} gelu(centers[c,bb,l,:]·W_back[:,m] + b_back[m])
// ---------------------------------------------------------------------------
__global__ __launch_bounds__(256) void output_kernel(
    const float* __restrict__ Wb, const float* __restrict__ bbk,
    const float* __restrict__ ws, float* __restrict__ out)
{
    int gid = blockIdx.x * 256 + threadIdx.x;   // < 1048576
    int d  = gid & 63;
    int l2 = (gid >> 6) & 255;
    int h  = (gid >> 14) & 7;
    int bp = gid >> 17;
    int m = ((l2 & 7) << 6) | d;
    int tbase = ((bp * 8 + h) * 5) * 32 + (l2 >> 3);

    float wcol[5];
    #pragma unroll
    for (int j = 0; j < 5; ++j) wcol[j] = Wb[j * DM + m];
    float bias = bbk[m];
    const float* ct = ws + CTOFF;

    float best = -INFINITY;
    #pragma unroll
    for (int cp = 0; cp < 5; ++cp) {
        int t = tbase + cp * 32;
        int l = t & 255;
        int t2 = t >> 8;
        int b2 = t2 & 7;
        int c  = t2 >> 3;
        const float* ctr = ct + ((c * BB + b2) * LL + l) * 5;
        float s = bias;
        #pragma unroll
        for (int j = 0; j < 5; ++j) s = fmaf(ctr[j], wcol[j], s);
        best = fmaxf(best, gelu_exact(s));
    }
    out[gid] = best;
}

// ---------------------------------------------------------------------------
// Loss kernels
// ---------------------------------------------------------------------------
__global__ void init_acc_kernel(float* ws) {
    ws[ACCOFF] = 0.0f; ws[ACCOFF + 1] = 0.0f;
}

// CE over the L axis: one block per (b,u) column of mu
__global__ __launch_bounds__(256) void ce_kernel(float* __restrict__ ws)
{
    __shared__ float red[256];
    int p = blockIdx.x;                 // 0..399
    int bb = p / UU;
    int u  = p - bb * UU;
    int l  = threadIdx.x;
    float v = ws[MUOFF + bb * ROWS_PER_B + l * UU + u];
    red[l] = v; __syncthreads();
    for (int s = 128; s > 0; s >>= 1) {
        if (l < s) red[l] = fmaxf(red[l], red[l + s]);
        __syncthreads();
    }
    float mx = red[0]; __syncthreads();
    float e = expf(v - mx);
    red[l] = e; __syncthreads();
    for (int s = 128; s > 0; s >>= 1) {
        if (l < s) red[l] += red[l + s];
        __syncthreads();
    }
    float Ssum = red[0]; __syncthreads();
    float contrib = -v * (v - mx - logf(Ssum));
    red[l] = contrib; __syncthreads();
    for (int s = 128; s > 0; s >>= 1) {
        if (l < s) red[l] += red[l + s];
        __syncthreads();
    }
    if (l == 0) atomicAdd(&ws[ACCOFF + 1], red[0]);
}

__global__ __launch_bounds__(256) void lp_sum_kernel(float* __restrict__ ws)
{
    __shared__ float red[256];
    int gid = blockIdx.x * 256 + threadIdx.x;   // < 102400
    red[threadIdx.x] = ws[LPOFF + gid];
    __syncthreads();
    for (int s = 128; s > 0; s >>= 1) {
        if (threadIdx.x < s) red[threadIdx.x] += red[threadIdx.x + s];
        __syncthreads();
    }
    if (threadIdx.x == 0) atomicAdd(&ws[ACCOFF], red[0]);
}

__global__ void finalize_kernel(const float* __restrict__ ws, float* __restrict__ out)
{
    out[OUT_ELTS] = -(ws[ACCOFF] / (float)NROWS) + ws[ACCOFF + 1] / (float)(BB * UU);
}

// ---------------------------------------------------------------------------
extern "C" void kernel_launch(void* const* d_in, const int* in_sizes, int n_in,
                              void* d_out, int out_size, void* d_ws, size_t ws_size,
                              hipStream_t stream)
{
    const float* K    = (const float*)d_in[0];
    // d_in[1] = V (unused by reference)
    const float* Wk   = (const float*)d_in[2];
    const float* bk   = (const float*)d_in[3];
    const float* Wq   = (const float*)d_in[4];
    const float* bq   = (const float*)d_in[5];
    const float* Wbk  = (const float*)d_in[6];
    const float* bbk  = (const float*)d_in[7];
    float* out = (float*)d_out;
    float* ws  = (float*)d_ws;

    // 6400 tiles of 16 rows, 4 waves per 128-thread block
    cluster_kernel<<<1600, 128, 0, stream>>>(K, Wk, bk, Wq, bq, ws);

    // loss path
    init_acc_kernel<<<1, 1, 0, stream>>>(ws);
    ce_kernel<<<BB * UU, 256, 0, stream>>>(ws);
    lp_sum_kernel<<<NROWS / 256, 256, 0, stream>>>(ws);
    finalize_kernel<<<1, 1, 0, stream>>>(ws, out);

    // output path
    centers_kernel<<<(BB * LL + 255) / 256, 256, 0, stream>>>(ws);
    output_kernel<<<OUT_ELTS / 256, 256, 0, stream>>>(Wbk, bbk, ws, out);
}